// SinusoidalEmbedding_83030307766274
// MI455X (gfx1250) — compile-verified
//
#include <hip/hip_runtime.h>

// SinusoidalEmbedding for MI455X (gfx1250, wave32).
//   out[b,s,d] = emb_table[x[b,s], d] + pe[s,d]
// B=8, S=4096, D=1024, fp32. Pure bandwidth kernel: ~256 MiB HBM traffic,
// roofline ~11 us at 23.3 TB/s. Strategy:
//   * one block per sequence position s; PE (sin/cos) computed ONCE and
//     reused for all 8 batch rows (8x fewer transcendental ops)
//   * gathered table rows streamed through a 4-deep LDS ring with gfx1250
//     global_load_async_to_lds_b128 (ASYNCcnt): 4 random-row fetches in
//     flight per thread, overlapping HBM latency with the PE add
//   * output written with nontemporal b128 stores so the 128 MB write-once
//     output does not evict the ~196 MB table from the 192 MB L2

typedef float v4f __attribute__((ext_vector_type(4)));

#define SEQ    4096
#define DM     1024
#define NBATCH 8
#define BUFS   4       // async pipeline depth (ring of 4 KB row buffers)

#if defined(__has_builtin) && __has_builtin(__builtin_amdgcn_s_wait_asynccnt)
#define WAIT_ASYNC(n) __builtin_amdgcn_s_wait_asynccnt(n)
#else
#define WAIT_ASYNC(n) asm volatile("s_wait_asynccnt %0" :: "i"(n) : "memory")
#endif

// Async DMA of 16 bytes/lane from global memory into this lane's LDS slot.
#define ASYNC_LOAD_B128(ldsaddr, gaddr)                               \
  asm volatile("global_load_async_to_lds_b128 %0, %1, off"            \
               :: "v"(ldsaddr), "v"(gaddr) : "memory")

// Same, but carries a dummy register dependency (%2) so the re-issue into a
// recycled ring slot is ordered after the consuming ds_load's result.
#define ASYNC_LOAD_B128_DEP(ldsaddr, gaddr, dep)                      \
  asm volatile("global_load_async_to_lds_b128 %0, %1, off"            \
               :: "v"(ldsaddr), "v"(gaddr), "v"(dep) : "memory")

__global__ __launch_bounds__(256)
void SinusoidalEmbedding_gfx1250_kernel(const int* __restrict__ x,
                                        const float* __restrict__ emb,
                                        float* __restrict__ out) {
  __shared__ __align__(16) float buf[BUFS][DM];   // 16 KB ring buffer

  const int s  = blockIdx.x;      // sequence position, 0..4095
  const int t  = threadIdx.x;     // 0..255
  const int d0 = t << 2;          // this lane owns d0..d0+3

  // ---- positional encoding, computed once, reused for all 8 batches ----
  // inv_freq(i) = 10000^(-2i/D) = exp2(-i * 2*log2(10000)/D)
  const float kC = -2.0f * 13.287712379549449f / (float)DM;
  const float f0 = exp2f((float)(2 * t)     * kC);
  const float f1 = exp2f((float)(2 * t + 1) * kC);
  const float a0 = (float)s * f0;
  const float a1 = (float)s * f1;
  v4f pe;
  pe.x = __sinf(a0); pe.y = __cosf(a0);
  pe.z = __sinf(a1); pe.w = __cosf(a1);

  // ---- indices: uniform per block -> scalar loads ----
  int idx[NBATCH];
#pragma unroll
  for (int b = 0; b < NBATCH; ++b) idx[b] = x[b * SEQ + s];

  // LDS byte addresses of this lane's 16B slot in each ring buffer
  // (low 32 bits of a flat LDS pointer == LDS byte offset per ISA 10.2)
  unsigned ldsA[BUFS];
#pragma unroll
  for (int w = 0; w < BUFS; ++w)
    ldsA[w] = (unsigned)(unsigned long long)(&buf[w][d0]);

  // ---- prologue: fill the async pipeline (4 rows in flight) ----
#pragma unroll
  for (int b = 0; b < BUFS && b < NBATCH; ++b) {
    unsigned long long ga =
        (unsigned long long)(emb + (size_t)idx[b] * DM + d0);
    ASYNC_LOAD_B128(ldsA[b & (BUFS - 1)], ga);
  }

  // ---- steady state ----
#pragma unroll
  for (int b = 0; b < NBATCH; ++b) {
    // async ops complete in order: after this wait, row b has landed.
    // outstanding-after-wait = min(BUFS-1, rows still in flight)
    constexpr int dummy = 0; (void)dummy;
    const int remain = NBATCH - 1 - b;
    if (remain >= BUFS - 1) { WAIT_ASYNC(BUFS - 1); }
    else if (remain == 2)   { WAIT_ASYNC(2); }
    else if (remain == 1)   { WAIT_ASYNC(1); }
    else                    { WAIT_ASYNC(0); }

    // ds_load_b128 of this lane's own slot (no cross-lane use -> no barrier)
    v4f v = *(const v4f*)(&buf[b & (BUFS - 1)][d0]);
    v.x += pe.x; v.y += pe.y; v.z += pe.z; v.w += pe.w;

    // refill this ring slot with row b+BUFS; dep on v.x orders the async
    // write after the ds_load's data has returned (WAR on the slot)
    if (b + BUFS < NBATCH) {
      unsigned long long ga =
          (unsigned long long)(emb + (size_t)idx[b + BUFS] * DM + d0);
      ASYNC_LOAD_B128_DEP(ldsA[b & (BUFS - 1)], ga, v.x);
    }

    // streaming store: keep L2 for the embedding table
    float* op = out + ((size_t)b * SEQ + s) * DM + d0;
    __builtin_nontemporal_store(v, (v4f*)op);
  }
}

extern "C" void kernel_launch(void* const* d_in, const int* in_sizes, int n_in,
                              void* d_out, int out_size, void* d_ws, size_t ws_size,
                              hipStream_t stream) {
  (void)in_sizes; (void)n_in; (void)out_size; (void)d_ws; (void)ws_size;
  const int*   x   = (const int*)d_in[0];    // [8, 4096] token ids
  const float* emb = (const float*)d_in[1];  // [50257, 1024] fp32
  float*       out = (float*)d_out;          // [8, 4096, 1024] fp32

  dim3 grid(SEQ);     // one block per sequence position
  dim3 block(256);    // 8 wave32s; each lane owns 4 consecutive d's
  hipLaunchKernelGGL(SinusoidalEmbedding_gfx1250_kernel, grid, block, 0, stream,
                     x, emb, out);
}